// waveletDecomp_23021024707188
// MI455X (gfx1250) — compile-verified
//
#include <hip/hip_runtime.h>
#include <hip/hip_bf16.h>

typedef __attribute__((ext_vector_type(2))) float v2f;
typedef __attribute__((ext_vector_type(8))) float v8f;

// x:   (8, 64, 512, 512) fp32, NCHW
// out: (8, 256, 256, 256) fp32; out channel = c*4 + f, f in {LL, LH, HL, HH}
//
// One wave32 handles one full output row (n, c, oi, 0..255) = 16 WMMA tiles.
// Phase 1: all 16 B-operand loads issued as one clause (16 in flight).
// Phase 2: 16x v_wmma_f32_16x16x4_f32 + clause'd stores.
//
//   B (4x16): K0/K1 = lanes 0-15 (.x/.y of even-row float2 load)
//             K2/K3 = lanes 16-31 (.x/.y of odd-row float2 load)
//   A (16x4): rows 0-3 = filters 0-3 ; rows 8,9 = filters 2,3 (others 0)
//   => D vgpr0 = f0 (lanes 0-15) | f2 (lanes 16-31)
//      D vgpr1 = f1 (lanes 0-15) | f3 (lanes 16-31)
__global__ __launch_bounds__(256) void haar_wmma_kernel(const float* __restrict__ x,
                                                        float* __restrict__ out) {
    const int lane   = threadIdx.x & 31;
    const int warp   = threadIdx.x >> 5;
    const int wg     = blockIdx.x * 8 + warp;   // global warp id: one per output row
    const int oi     = wg & 255;                // output row 0..255
    const int nc     = wg >> 8;                 // n*64 + c, 0..511
    const int rowSel = lane >> 4;               // 0: even input row, 1: odd input row
    const int m      = lane & 15;               // block index within tile / output col lsb

    // ---- A operand: 16x4 fp32 Haar matrix (scale 0.25 folds filter + /2) ----
    // filter f coefficient signs vs (p00,p01,p10,p11):
    //   f0:{+,+,+,+} f1:{+,+,-,-} f2:{+,-,+,-} f3:{+,-,-,+}
    const int f = (m < 4) ? m : ((m == 8) ? 2 : ((m == 9) ? 3 : -1));
    const float s   = (f >= 0) ? 0.25f : 0.0f;
    const float sg1 = (f & 2) ? -1.0f : 1.0f;   // sign at k=1 (p01)
    const float sg2 = (f & 1) ? -1.0f : 1.0f;   // sign at k=2 (p10)
    const float sg3 = sg1 * sg2;                // sign at k=3 (p11)
    v2f A;
    A.x = rowSel ? s * sg2 : s;        // vgpr0: lanes<16 -> k=0, lanes>=16 -> k=2
    A.y = rowSel ? s * sg3 : s * sg1;  // vgpr1: lanes<16 -> k=1, lanes>=16 -> k=3

    // ---- addresses ----
    // input row base: plane nc, input row 2*oi + rowSel, col = 2*m + 32*jt
    const float* rowPtr = x + ((long long)nc * 512 + 2 * oi + rowSel) * 512 + 2 * m;
    // output: lanes 0-15 write filters {0,1}; lanes 16-31 write filters {2,3}
    const long long outBase0 =
        ((long long)(nc * 4 + (rowSel ? 2 : 0)) * 256 + oi) * 256;  // f0 / f2 plane row
    const long long outBase1 = outBase0 + 256LL * 256LL;            // f1 / f3 plane row

    // Phase 1: issue all 16 tile loads back-to-back (one clause, 16 in flight).
    // Streams have zero reuse -> non-temporal so we don't pollute the 192MB L2.
    v2f Bt[16];
#pragma unroll
    for (int jt = 0; jt < 16; ++jt)
        Bt[jt] = __builtin_nontemporal_load((const v2f*)(rowPtr + jt * 32));

    // Phase 2: 16 WMMAs; waits degrade gracefully (loadcnt 15..0).
#pragma unroll
    for (int jt = 0; jt < 16; ++jt) {
        v8f C = {};
        v8f D = __builtin_amdgcn_wmma_f32_16x16x4_f32(
            /*neg_a=*/false, A, /*neg_b=*/false, Bt[jt],
            /*c_mod=*/(short)0, C, /*reuse_a=*/false, /*reuse_b=*/false);

        const int oj = jt * 16 + m;
        __builtin_nontemporal_store(D[0], out + outBase0 + oj);  // f0 | f2
        __builtin_nontemporal_store(D[1], out + outBase1 + oj);  // f1 | f3
    }
}

extern "C" void kernel_launch(void* const* d_in, const int* in_sizes, int n_in,
                              void* d_out, int out_size, void* d_ws, size_t ws_size,
                              hipStream_t stream) {
    (void)in_sizes; (void)n_in; (void)out_size; (void)d_ws; (void)ws_size;
    const float* x = (const float*)d_in[0];
    float* out = (float*)d_out;
    // 512 (n*c) planes * 256 output rows = 131072 warps; 8 warps/block
    dim3 grid(16384), block(256);
    hipLaunchKernelGGL(haar_wmma_kernel, grid, block, 0, stream, x, out);
}